// ResNet1_18038862643493
// MI455X (gfx1250) — compile-verified
//
#include <hip/hip_runtime.h>

// ---------------------------------------------------------------------------
// Neural-ODE RK4 on MI455X (gfx1250), wave32.
// 16 samples per wave ride the M dimension of V_WMMA_F32_16X16X4_F32.
// Layer1 (K=2 padded to 4): 8 WMMAs over 8 hidden tiles, W1 resident in VGPRs.
// tanh in VALU, activations bounced through per-wave LDS (stride-padded) to
// transpose C-layout -> A-layout. Layer2 (K=128): 32 chained WMMAs in 4
// independent accumulator chains. All sync is intra-wave via s_wait_dscnt.
// ---------------------------------------------------------------------------

typedef __attribute__((ext_vector_type(2))) float v2f;
typedef __attribute__((ext_vector_type(8))) float v8f;

#define HID      128
#define NT       8      // hidden tiles (128/16) for layer 1 N-dim
#define KT       32     // K tiles (128/4) for layer 2
#define WAVES    4      // waves per block
#define SPW      16     // samples per wave (M of the WMMA tile)
#define HSTRIDE  132    // padded dword stride for activation buffer (bank-safe)
#define STEPH    0.01f

__device__ __forceinline__ float fast_tanh(float x) {
#if __has_builtin(__builtin_amdgcn_tanhf)
  return __builtin_amdgcn_tanhf(x);            // v_tanh_f32 (TRANS)
#else
  // tanh(x) = 1 - 2/(exp2(2x*log2e)+1): v_exp_f32 + v_rcp_f32
  float e = __builtin_amdgcn_exp2f(x * 2.8853900817779268f);
  return 1.0f - 2.0f * __builtin_amdgcn_rcpf(e + 1.0f);
#endif
}

__device__ __forceinline__ void wait_lds() {
  // LDS ops from the same wave complete in order; DScnt==0 => stores visible,
  // loads landed. asm volatile + memory clobber also pins compiler ordering.
  asm volatile("s_wait_dscnt 0" ::: "memory");
}

__device__ __forceinline__ v8f wmma4(v2f a, v2f b, v8f c) {
  // (neg_a, A, neg_b, B, c_mod, C, reuse_a, reuse_b)
  return __builtin_amdgcn_wmma_f32_16x16x4_f32(false, a, false, b, (short)0, c,
                                               false, false);
}

__global__ __launch_bounds__(32 * WAVES, 1)
void node_rk4_wmma(const float* __restrict__ x,
                   const float* __restrict__ W1, const float* __restrict__ b1,
                   const float* __restrict__ W2, const float* __restrict__ b2,
                   const float* __restrict__ Wf, const float* __restrict__ bf,
                   const int* __restrict__ tptr,
                   float* __restrict__ out, int n) {
  __shared__ float ldsH[WAVES][SPW * HSTRIDE];  // activations, per-wave slice
  __shared__ float ldsY[WAVES][SPW * 2];        // 16x2 k-transpose scratch

  const int  tid  = threadIdx.x;
  const int  wave = tid >> 5;
  const int  lane = tid & 31;
  const int  lm   = lane & 15;       // N (or M) index within a 16-wide tile
  const bool hi   = lane >= 16;      // upper half-wave: K+2/K+3 (A/B), M+8 (C)
  const int  M0   = hi ? 8 : 0;

  // ---- resident weights in WMMA B-layout -----------------------------------
  // B 4x16 f32: v.x = row K (lo lanes) / K+2 (hi lanes); v.y = K+1 / K+3.
  v2f   w1b[NT];
  float b1v[NT];
#pragma unroll
  for (int t = 0; t < NT; ++t) {
    w1b[t].x = hi ? 0.0f : W1[0 * HID + 16 * t + lm];  // K=0 row; K=2 pad = 0
    w1b[t].y = hi ? 0.0f : W1[1 * HID + 16 * t + lm];  // K=1 row; K=3 pad = 0
    b1v[t]   = b1[16 * t + lm];                        // bias depends on N only
  }
  v2f w2b[KT];
#pragma unroll
  for (int k = 0; k < KT; ++k) {
    int k0   = 4 * k + (hi ? 2 : 0);
    w2b[k].x = (lm < 2) ? W2[k0 * 2 + lm]       : 0.0f;
    w2b[k].y = (lm < 2) ? W2[(k0 + 1) * 2 + lm] : 0.0f;
  }
  const float b2v = (lm < 2) ? b2[lm] : 0.0f;

  // ---- initial state, A-layout (lane m holds sample m; hi lanes are K-pad) -
  const int base = (blockIdx.x * WAVES + wave) * SPW;
  const int s    = base + lm;
  const int sc   = (s < n) ? s : (n - 1);
  v2f y;
  y.x = hi ? 0.0f : x[2 * sc + 0];
  y.y = hi ? 0.0f : x[2 * sc + 1];

  float* Hbuf = ldsH[wave];
  float* Ybuf = ldsY[wave];

  // ---- one evaluation of k = h * g(y); y given & returned in A-layout ------
  auto geval = [&](v2f yin) -> v2f {
    wait_lds();  // previous stage's loads of Hbuf fully retired
    // layer 1: 8 tiles of 16 hidden units, then tanh -> LDS (C -> A transpose)
#pragma unroll
    for (int t = 0; t < NT; ++t) {
      v8f c;
#pragma unroll
      for (int r = 0; r < 8; ++r) c[r] = b1v[t];
      c = wmma4(yin, w1b[t], c);
      const int col = 16 * t + lm;
#pragma unroll
      for (int r = 0; r < 8; ++r)
        Hbuf[(M0 + r) * HSTRIDE + col] = fast_tanh(c[r]);
    }
    wait_lds();
    // reload as A-layout K-tiles: lane m, tile k -> H[m][4k + (hi?2:0)] pair
    v2f a[KT];
#pragma unroll
    for (int k = 0; k < KT; ++k) {
      const int off = lm * HSTRIDE + 4 * k + (hi ? 2 : 0);  // 8B aligned
      a[k].x = Hbuf[off];
      a[k].y = Hbuf[off + 1];
    }
    // layer 2: K=128 via 32 WMMAs, 4 independent accumulator chains
    v8f a0, a1, a2, a3;
#pragma unroll
    for (int r = 0; r < 8; ++r) { a0[r] = b2v; a1[r] = 0.f; a2[r] = 0.f; a3[r] = 0.f; }
#pragma unroll
    for (int k = 0; k < KT; k += 4) {
      a0 = wmma4(a[k + 0], w2b[k + 0], a0);
      a1 = wmma4(a[k + 1], w2b[k + 1], a1);
      a2 = wmma4(a[k + 2], w2b[k + 2], a2);
      a3 = wmma4(a[k + 3], w2b[k + 3], a3);
    }
#pragma unroll
    for (int r = 0; r < 8; ++r) a0[r] = (a0[r] + a1[r]) + (a2[r] + a3[r]);
    // 16x2 result: C-layout -> A-layout via tiny LDS bounce, scaled by h
    wait_lds();
    if (lm < 2) {
#pragma unroll
      for (int r = 0; r < 8; ++r) Ybuf[(M0 + r) * 2 + lm] = STEPH * a0[r];
    }
    wait_lds();
    v2f kk; kk.x = 0.0f; kk.y = 0.0f;
    if (!hi) { kk.x = Ybuf[lm * 2 + 0]; kk.y = Ybuf[lm * 2 + 1]; }
    return kk;
  };

  // ---- RK4 time loop (t python int -> 10*t + 1 steps) ----------------------
  const int nsteps = tptr[0] * 10 + 1;
#pragma unroll 1
  for (int it = 0; it < nsteps; ++it) {
    v2f k1 = geval(y);
    v2f t2; t2.x = y.x + 0.5f * k1.x; t2.y = y.y + 0.5f * k1.y;
    v2f k2 = geval(t2);
    v2f t3; t3.x = y.x + 0.5f * k2.x; t3.y = y.y + 0.5f * k2.y;
    v2f k3 = geval(t3);
    v2f t4; t4.x = y.x + k3.x; t4.y = y.y + k3.y;
    v2f k4 = geval(t4);
    y.x += (k1.x + 2.0f * k2.x + 2.0f * k3.x + k4.x) * (1.0f / 6.0f);
    y.y += (k1.y + 2.0f * k2.y + 2.0f * k3.y + k4.y) * (1.0f / 6.0f);
  }

  // ---- head: logits = y @ Wf + bf, softmax; out = [logits | probas] --------
  if (!hi && s < n) {
    const float l0 = y.x * Wf[0] + y.y * Wf[2] + bf[0];
    const float l1 = y.x * Wf[1] + y.y * Wf[3] + bf[1];
    const float mx = fmaxf(l0, l1);
    const float e0 = __builtin_amdgcn_exp2f((l0 - mx) * 1.4426950408889634f);
    const float e1 = __builtin_amdgcn_exp2f((l1 - mx) * 1.4426950408889634f);
    const float iv = __builtin_amdgcn_rcpf(e0 + e1);
    out[2 * (size_t)s + 0] = l0;
    out[2 * (size_t)s + 1] = l1;
    out[2 * (size_t)n + 2 * (size_t)s + 0] = e0 * iv;
    out[2 * (size_t)n + 2 * (size_t)s + 1] = e1 * iv;
  }
}

extern "C" void kernel_launch(void* const* d_in, const int* in_sizes, int n_in,
                              void* d_out, int out_size, void* d_ws, size_t ws_size,
                              hipStream_t stream) {
  (void)n_in; (void)out_size; (void)d_ws; (void)ws_size;
  const float* x  = (const float*)d_in[0];
  const float* W1 = (const float*)d_in[1];
  const float* b1 = (const float*)d_in[2];
  const float* W2 = (const float*)d_in[3];
  const float* b2 = (const float*)d_in[4];
  const float* Wf = (const float*)d_in[5];
  const float* bf = (const float*)d_in[6];
  const int*   t  = (const int*)d_in[7];
  const int n = in_sizes[0] / 2;                 // N samples (x is N x 2)
  const int spb = WAVES * SPW;                   // 64 samples per block
  const int blocks = (n + spb - 1) / spb;
  node_rk4_wmma<<<blocks, 32 * WAVES, 0, stream>>>(x, W1, b1, W2, b2, Wf, bf, t,
                                                   (float*)d_out, n);
}